// GDG_34557306863694
// MI455X (gfx1250) — compile-verified
//
#include <hip/hip_runtime.h>
#include <hip/hip_bf16.h>
#include <float.h>

typedef __attribute__((ext_vector_type(2))) float v2f;
typedef __attribute__((ext_vector_type(8))) float v8f;
typedef __attribute__((ext_vector_type(4))) unsigned int u32x4;
typedef __attribute__((ext_vector_type(8))) int i32x8;
typedef __attribute__((ext_vector_type(4))) int i32x4;

#if defined(__has_builtin)
#  if __has_builtin(__builtin_amdgcn_tensor_load_to_lds) && __has_builtin(__builtin_amdgcn_s_wait_tensorcnt)
#    define USE_TDM 1
#  endif
#endif
#ifndef USE_TDM
#  define USE_TDM 0
#endif

// Problem constants (from the reference)
constexpr int Bc = 128;   // batches
constexpr int Nc = 196;   // points per batch
constexpr int Dc = 1024;  // feature dim
constexpr int Kc = 8;     // neighbors kept

// Tiling: 14x14 grid of 16x16 WMMA tiles -> 7x7 grid of 32x32 wave blocks
constexpr int NT    = 14;          // 14 * 16 = 224 >= 196
constexpr int NPAD  = NT * 16;     // 224 padded rows/cols
constexpr int NB    = 7;           // 7 * 32 = 224 (2x2 register-blocked)
constexpr int GS    = 227;         // Gram LDS row stride (odd -> conflict-free column reads)
constexpr int KCHNK = 128;         // K chunk staged in LDS per iteration
constexpr int XS    = KCHNK + 1;   // X-chunk LDS row stride (odd; TDM pad gives this for free)
constexpr int GRAM_ELEMS = NPAD * GS;   // 224*227 = 50,848 f32 (203,392 B)
constexpr int X_ELEMS    = NPAD * XS;   // 224*129 = 28,896 f32 (115,584 B)
constexpr size_t SMEM_BYTES = (size_t)(GRAM_ELEMS + X_ELEMS) * sizeof(float); // 318,976 B <= 320 KB

#if USE_TDM
// Issue one TDM descriptor: 2D tile (KCHNK x Nc rows) of f32, global row pitch Dc,
// LDS destination padded to a 129-DWORD row stride via pad_interval/pad_amount.
__device__ __forceinline__ void tdm_load_chunk(const float* src, unsigned lds_byte_off)
{
    unsigned long long ga = (unsigned long long)(uintptr_t)src;
    u32x4 g0;
    g0.x = 1u;                                   // count=1 (valid user descriptor)
    g0.y = lds_byte_off;                         // lds_addr (bytes)
    g0.z = (unsigned)ga;                         // global_addr[31:0] (tile start)
    g0.w = (unsigned)((ga >> 32) & 0x01FFFFFFull)// global_addr[56:32]
         | (2u << 30);                           // type=2 ("image")
    i32x8 g1;
    g1[0] = (int)((2u << 16)                     // data_size = 4 bytes
                | (1u << 20)                     // pad_enable
                | (6u << 22)                     // pad_interval: after 128 DWORDs
                | (0u << 25));                   // pad_amount: 1 DWORD -> 129-DWORD LDS rows
    g1[1] = (int)(((unsigned)Dc & 0xFFFFu) << 16);                 // tensor_dim0[15:0]
    g1[2] = (int)(((unsigned)Dc >> 16) | (((unsigned)Nc) << 16));  // tensor_dim0[31:16] | tensor_dim1[15:0]
    g1[3] = (int)(((unsigned)Nc >> 16) | (((unsigned)KCHNK) << 16));// tensor_dim1[31:16] | tile_dim0
    g1[4] = (int)(unsigned)Nc;                   // tile_dim1 = 196 rows; tile_dim2 = 0 (2D)
    g1[5] = (int)(unsigned)Dc;                   // tensor_dim0_stride[31:0] (row pitch, elements)
    g1[6] = 0;                                   // stride hi | tensor_dim1_stride lo (unused, 2D)
    g1[7] = 0;
    i32x4 gz4 = {0, 0, 0, 0};
    i32x8 gz8 = {0, 0, 0, 0, 0, 0, 0, 0};
    // 6-arg toolchain form: (group0, group1, group2, group3, group4, cpol)
    __builtin_amdgcn_tensor_load_to_lds(g0, g1, gz4, gz4, gz8, 0);
}
#endif

__global__ __launch_bounds__(256)
void knn_gram_wmma_kernel(const float* __restrict__ x, float* __restrict__ out)
{
    extern __shared__ float smem[];
    float* g  = smem;                 // [NPAD][GS]  Gram / distance matrix
    float* xs = smem + GRAM_ELEMS;    // [NPAD][XS]  staged X chunk (reused for diag later)

    const int b    = blockIdx.x;
    const int tid  = threadIdx.x;
    const int lane = tid & 31;
    const int wave = tid >> 5;
    const float* __restrict__ xb = x + (size_t)b * Nc * Dc;

    // WMMA f32 16x16x4 per-lane fragment coordinates (ISA 7.12.2):
    // A/B: lanes 0-15 -> K={0,1}, lanes 16-31 -> K={2,3}; row/col = lane&15
    const int fr = lane & 15;          // fragment row (A) / col (B)
    const int fk = (lane >> 4) * 2;    // fragment K base
    // C/D: N = lane&15 ; M = vgpr + (lane>>4)*8
    const int cn  = lane & 15;
    const int cm0 = (lane >> 4) * 8;

#if USE_TDM
    // Zero the padded X rows once (TDM only ever writes rows 0..Nc-1).
    for (int i = tid; i < (NPAD - Nc) * XS; i += 256)
        xs[Nc * XS + i] = 0.0f;
    const unsigned xs_lds_off = (unsigned)(uintptr_t)xs;  // flat LDS addr[31:0] == LDS byte offset
#endif

    for (int k0 = 0; k0 < Dc; k0 += KCHNK) {
#if USE_TDM
        if (wave == 0) {
            tdm_load_chunk(xb + k0, xs_lds_off);
            __builtin_amdgcn_s_wait_tensorcnt(0);
        }
#else
        // Fallback staging: float4 global loads, scalar LDS stores; zero padded rows.
        for (int idx = tid; idx < NPAD * (KCHNK / 4); idx += 256) {
            const int row = idx / (KCHNK / 4);
            const int c4  = idx % (KCHNK / 4);
            float4 v4 = make_float4(0.0f, 0.0f, 0.0f, 0.0f);
            if (row < Nc)
                v4 = *(const float4*)(xb + (size_t)row * Dc + k0 + c4 * 4);
            float* dst = xs + row * XS + c4 * 4;
            dst[0] = v4.x; dst[1] = v4.y; dst[2] = v4.z; dst[3] = v4.w;
        }
#endif
        __syncthreads();

        // ---- Each wave accumulates 32x32 output blocks (2x2 WMMA tiles) ----
        for (int t = wave; t < NB * NB; t += 8) {
            const int rowA = (t / NB) * 32;
            const int rowB = (t % NB) * 32;

            v8f acc[2][2];
            float* cbase = g + (size_t)(rowA + cm0) * GS + rowB + cn;
            if (k0 == 0) {
                #pragma unroll
                for (int ii = 0; ii < 2; ++ii)
                    #pragma unroll
                    for (int jj = 0; jj < 2; ++jj)
                        #pragma unroll
                        for (int r = 0; r < 8; ++r) acc[ii][jj][r] = 0.0f;
            } else {
                #pragma unroll
                for (int ii = 0; ii < 2; ++ii)
                    #pragma unroll
                    for (int jj = 0; jj < 2; ++jj) {
                        float* cp = cbase + (size_t)ii * 16 * GS + jj * 16;
                        #pragma unroll
                        for (int r = 0; r < 8; ++r) acc[ii][jj][r] = cp[(size_t)r * GS];
                    }
            }

            const float* __restrict__ ap0 = xs + (rowA + fr) * XS + fk;
            const float* __restrict__ ap1 = ap0 + 16 * XS;
            const float* __restrict__ bp0 = xs + (rowB + fr) * XS + fk;
            const float* __restrict__ bp1 = bp0 + 16 * XS;

            #pragma unroll
            for (int kk = 0; kk < KCHNK; kk += 4) {
                v2f a0, a1, b0, b1;
                a0[0] = ap0[kk]; a0[1] = ap0[kk + 1];
                a1[0] = ap1[kk]; a1[1] = ap1[kk + 1];
                b0[0] = bp0[kk]; b0[1] = bp0[kk + 1];
                b1[0] = bp1[kk]; b1[1] = bp1[kk + 1];
                acc[0][0] = __builtin_amdgcn_wmma_f32_16x16x4_f32(false, a0, false, b0,
                                (short)0, acc[0][0], false, false);
                acc[0][1] = __builtin_amdgcn_wmma_f32_16x16x4_f32(false, a0, false, b1,
                                (short)0, acc[0][1], false, false);
                acc[1][0] = __builtin_amdgcn_wmma_f32_16x16x4_f32(false, a1, false, b0,
                                (short)0, acc[1][0], false, false);
                acc[1][1] = __builtin_amdgcn_wmma_f32_16x16x4_f32(false, a1, false, b1,
                                (short)0, acc[1][1], false, false);
            }

            #pragma unroll
            for (int ii = 0; ii < 2; ++ii)
                #pragma unroll
                for (int jj = 0; jj < 2; ++jj) {
                    float* cp = cbase + (size_t)ii * 16 * GS + jj * 16;
                    #pragma unroll
                    for (int r = 0; r < 8; ++r) cp[(size_t)r * GS] = acc[ii][jj][r];
                }
        }
        __syncthreads();
    }

    // ---- Gram complete: squared norms are the diagonal. Stage into xs. ----
    float* diag = xs;
    for (int m = tid; m < Nc; m += 256) diag[m] = g[(size_t)m * GS + m];
    __syncthreads();

    // ---- Per-row top-(K+1) smallest distances (stable: ties keep lower index) ----
    if (tid < Nc) {
        const int n = tid;
        const float sqn = diag[n];
        const float* __restrict__ grow = g + (size_t)n * GS;

        float bd[Kc + 1];
        int   bi[Kc + 1];
        #pragma unroll
        for (int j = 0; j <= Kc; ++j) { bd[j] = FLT_MAX; bi[j] = -1; }

        for (int m = 0; m < Nc; ++m) {
            const float d2 = sqn + diag[m] - 2.0f * grow[m];
            const float d  = sqrtf(fmaxf(d2, 0.0f));
            if (d < bd[Kc]) {
                int j = Kc;
                while (j > 0 && d < bd[j - 1]) {
                    bd[j] = bd[j - 1]; bi[j] = bi[j - 1]; --j;
                }
                bd[j] = d; bi[j] = m;
            }
        }

        // Entry 0 is self (distance exactly 0); emit entries 1..K.
        const size_t seg  = (size_t)Bc * Nc * Kc;
        const size_t base = ((size_t)b * Nc + n) * Kc;
        float* outD = out;              // knn_dists  [B,N,K]
        float* outN = out + seg;        // neighbors  [B,N*K] (as float values)
        float* outS = out + 2 * seg;    // src_index  [B,N*K] (as float values)
        #pragma unroll
        for (int j = 0; j < Kc; ++j) {
            outD[base + j] = bd[j + 1];
            outN[base + j] = (float)bi[j + 1];
            outS[base + j] = (float)n;
        }
    }
}

extern "C" void kernel_launch(void* const* d_in, const int* in_sizes, int n_in,
                              void* d_out, int out_size, void* d_ws, size_t ws_size,
                              hipStream_t stream) {
    (void)in_sizes; (void)n_in; (void)out_size; (void)d_ws; (void)ws_size;
    const float* xp = (const float*)d_in[0];
    float* outp = (float*)d_out;
    knn_gram_wmma_kernel<<<dim3(Bc), dim3(256), SMEM_BYTES, stream>>>(xp, outp);
}